// GATClassifier_2740189135670
// MI455X (gfx1250) — compile-verified
//
#include <hip/hip_runtime.h>
#include <hip/hip_bf16.h>
#include <cstddef>

// ---------------------------------------------------------------------------
// GATClassifier pipeline for MI455X (gfx1250, wave32).
// f32 end-to-end; matrix work on V_WMMA_F32_16X16X4_F32.
// ---------------------------------------------------------------------------

typedef __attribute__((ext_vector_type(2))) float v2f;
typedef __attribute__((ext_vector_type(8))) float v8f;

#define NEG_SLOPE 0.2f
#define GAT_EPS   1e-16f

// ---- monotone float<->uint ordering for atomicMax on floats ----------------
__device__ __forceinline__ unsigned ord_encode(float f) {
  unsigned u = __float_as_uint(f);
  return (u & 0x80000000u) ? ~u : (u | 0x80000000u);
}
__device__ __forceinline__ float ord_decode(unsigned u) {
  unsigned b = (u & 0x80000000u) ? (u & 0x7FFFFFFFu) : ~u;
  return __uint_as_float(b);
}

// ---------------------------------------------------------------------------
__global__ void fill_zero_kernel(float* __restrict__ p, size_t n) {
  size_t i = (size_t)blockIdx.x * blockDim.x + threadIdx.x;
  size_t stride = (size_t)gridDim.x * blockDim.x;
  for (; i < n; i += stride) p[i] = 0.0f;
}

// ---------------------------------------------------------------------------
// C[M,N] = A[M,K] @ B[K,N], all f32 row-major.
// Requires M % 16 == 0 and N % 16 == 0 (true for all calls here); K arbitrary.
// One 16x16 output tile per wave; 8 waves per 256-thread block.
//
// Lane layouts (ISA 7.12.2, 32-bit matrices, wave32):
//   A 16x4 : lanes 0-15 row M=lane  {VGPR0=K0, VGPR1=K1}
//            lanes16-31 row M=lane-16 {VGPR0=K2, VGPR1=K3}
//   B 4x16 : lanes 0-15 col N=lane  {VGPR0=K0, VGPR1=K1}
//            lanes16-31 col N=lane-16 {VGPR0=K2, VGPR1=K3}
//   C 16x16: VGPR r -> row r (lanes 0-15) / row r+8 (lanes 16-31), col = lane%16
// ---------------------------------------------------------------------------
__global__ void gemm_f32_wmma_kernel(const float* __restrict__ A,
                                     const float* __restrict__ B,
                                     float* __restrict__ C,
                                     int M, int N, int K) {
  const int wave = threadIdx.x >> 5;
  const int lane = threadIdx.x & 31;
  const int half = lane >> 4;   // 0: K pair {0,1}; 1: K pair {2,3}
  const int lr   = lane & 15;

  const int n_tiles = N >> 4;
  const int n_tile  = blockIdx.y * (blockDim.x >> 5) + wave;
  if (n_tile >= n_tiles) return;          // whole wave exits together (EXEC all-1 for WMMA)

  const int row0 = blockIdx.x << 4;
  const int col0 = n_tile << 4;

  const float* Arow = A + (size_t)(row0 + lr) * K;
  const float* Bcol = B + col0 + lr;

  v8f acc = {};
  const int Kfull = K & ~3;
  int k0 = 0;
  for (; k0 < Kfull; k0 += 4) {
    const int ka = k0 + (half << 1);
    v2f a, b;
    a.x = Arow[ka];
    a.y = Arow[ka + 1];
    b.x = Bcol[(size_t)ka * N];
    b.y = Bcol[(size_t)(ka + 1) * N];
    acc = __builtin_amdgcn_wmma_f32_16x16x4_f32(false, a, false, b,
                                                (short)0, acc, false, false);
  }
  if (k0 < K) {  // guarded tail, zero-padded K
    const int ka = k0 + (half << 1);
    v2f a, b;
    a.x = (ka     < K) ? Arow[ka]                   : 0.0f;
    a.y = (ka + 1 < K) ? Arow[ka + 1]               : 0.0f;
    b.x = (ka     < K) ? Bcol[(size_t)ka * N]       : 0.0f;
    b.y = (ka + 1 < K) ? Bcol[(size_t)(ka + 1) * N] : 0.0f;
    acc = __builtin_amdgcn_wmma_f32_16x16x4_f32(false, a, false, b,
                                                (short)0, acc, false, false);
  }

  float* Crow = C + (size_t)(row0 + (half << 3)) * N + col0 + lr;
#pragma unroll
  for (int r = 0; r < 8; ++r)
    Crow[(size_t)r * N] = acc[r];
}

// ---------------------------------------------------------------------------
// al_src[n,h] = dot(H[n,h,:], a_src[h,:]); likewise al_dst.  One wave per (n,h).
// ---------------------------------------------------------------------------
__global__ void attn_scores_kernel(const float* __restrict__ Hfeat,
                                   const float* __restrict__ a_src,
                                   const float* __restrict__ a_dst,
                                   float* __restrict__ als,
                                   float* __restrict__ ald,
                                   int N, int NH, int C) {
  const int gw = blockIdx.x * (blockDim.x >> 5) + (threadIdx.x >> 5);
  if (gw >= N * NH) return;
  const int lane = threadIdx.x & 31;
  const int h = gw % NH;
  const float* hp = Hfeat + (size_t)gw * C;   // [N,NH,C] contiguous
  const float* as = a_src + (size_t)h * C;
  const float* ad = a_dst + (size_t)h * C;
  float ss = 0.f, sd = 0.f;
  for (int c = lane; c < C; c += 32) {
    float v = hp[c];
    ss += v * as[c];
    sd += v * ad[c];
  }
#pragma unroll
  for (int off = 16; off > 0; off >>= 1) {
    ss += __shfl_xor(ss, off, 32);
    sd += __shfl_xor(sd, off, 32);
  }
  if (lane == 0) { als[gw] = ss; ald[gw] = sd; }
}

// ---------------------------------------------------------------------------
// e = leaky_relu(al_s[src]+al_d[dst]); store e; segment-max into m[dst,h].
// Edge id >= E is the implicit self-loop (src=dst=e-E).
// ---------------------------------------------------------------------------
__global__ void edge_score_max_kernel(const int* __restrict__ srcA,
                                      const int* __restrict__ dstA,
                                      const float* __restrict__ als,
                                      const float* __restrict__ ald,
                                      float* __restrict__ ebuf,
                                      unsigned* __restrict__ mmax,
                                      int E, int ET, int NH) {
  int idx = blockIdx.x * blockDim.x + threadIdx.x;
  if (idx >= ET * NH) return;
  int e = idx / NH, h = idx - e * NH;
  int s, d;
  if (e < E) { s = srcA[e]; d = dstA[e]; } else { s = d = e - E; }
  float v = als[s * NH + h] + ald[d * NH + h];
  v = (v > 0.f) ? v : NEG_SLOPE * v;
  ebuf[idx] = v;
  atomicMax(&mmax[d * NH + h], ord_encode(v));
}

// ---------------------------------------------------------------------------
// ex = exp(e - m[dst]); store in place; segment-sum into denom[dst,h].
// ---------------------------------------------------------------------------
__global__ void edge_exp_sum_kernel(const int* __restrict__ dstA,
                                    float* __restrict__ ebuf,
                                    const unsigned* __restrict__ mmax,
                                    float* __restrict__ denom,
                                    int E, int ET, int NH) {
  int idx = blockIdx.x * blockDim.x + threadIdx.x;
  if (idx >= ET * NH) return;
  int e = idx / NH, h = idx - e * NH;
  int d = (e < E) ? dstA[e] : (e - E);
  float ex = expf(ebuf[idx] - ord_decode(mmax[d * NH + h]));
  ebuf[idx] = ex;
  atomicAdd(&denom[d * NH + h], ex);
}

// ---------------------------------------------------------------------------
// out[dst, :] += H[src, :] * alpha   (alpha per head).  One wave per edge.
// C == (1 << c_shift) features per head.
// ---------------------------------------------------------------------------
__global__ void edge_aggregate_kernel(const int* __restrict__ srcA,
                                      const int* __restrict__ dstA,
                                      const float* __restrict__ Hfeat,
                                      const float* __restrict__ exbuf,
                                      const float* __restrict__ denom,
                                      float* __restrict__ out,
                                      int E, int ET, int NH, int c_shift) {
  int gw = blockIdx.x * (blockDim.x >> 5) + (threadIdx.x >> 5);
  if (gw >= ET) return;
  int lane = threadIdx.x & 31;
  int s, d;
  if (gw < E) { s = srcA[gw]; d = dstA[gw]; } else { s = d = gw - E; }
  const int F = NH << c_shift;
  const float* hs = Hfeat + (size_t)s * F;
  float*       od = out   + (size_t)d * F;
  for (int f = lane; f < F; f += 32) {
    int h = f >> c_shift;
    float alpha = exbuf[gw * NH + h] / (denom[d * NH + h] + GAT_EPS);
    atomicAdd(&od[f], hs[f] * alpha);
  }
}

// ---------------------------------------------------------------------------
// out = act(in + bias[f]); mode 0 = ELU, mode 1 = ReLU.
// ---------------------------------------------------------------------------
__global__ void bias_act_kernel(const float* __restrict__ in,
                                const float* __restrict__ bias,
                                float* __restrict__ out,
                                size_t total, int F, int mode) {
  size_t i = (size_t)blockIdx.x * blockDim.x + threadIdx.x;
  size_t stride = (size_t)gridDim.x * blockDim.x;
  for (; i < total; i += stride) {
    float x = in[i] + bias[i % (size_t)F];
    out[i] = (mode == 0) ? ((x > 0.f) ? x : (expf(x) - 1.f))
                         : fmaxf(x, 0.f);
  }
}

// ---------------------------------------------------------------------------
// Final tiny GEMM: out[n,o] = sum_k r3[n,k]*W4[k,o] + b4[o]  (K=64, O=4).
// ---------------------------------------------------------------------------
__global__ void head_out_kernel(const float* __restrict__ r3,
                                const float* __restrict__ W4,
                                const float* __restrict__ b4,
                                float* __restrict__ out,
                                int N, int Kd, int Od) {
  int idx = blockIdx.x * blockDim.x + threadIdx.x;
  if (idx >= N * Od) return;
  int n = idx / Od, o = idx - n * Od;
  const float* r = r3 + (size_t)n * Kd;
  float s = b4[o];
  for (int k = 0; k < Kd; ++k) s += r[k] * W4[k * Od + o];
  out[idx] = s;
}

// ---------------------------------------------------------------------------
extern "C" void kernel_launch(void* const* d_in, const int* in_sizes, int n_in,
                              void* d_out, int out_size, void* d_ws, size_t ws_size,
                              hipStream_t stream) {
  const float* x   = (const float*)d_in[0];
  const int*   ei  = (const int*)  d_in[1];
  const float* W1  = (const float*)d_in[2];
  const float* as1 = (const float*)d_in[3];
  const float* ad1 = (const float*)d_in[4];
  const float* b1  = (const float*)d_in[5];
  const float* W2  = (const float*)d_in[6];
  const float* as2 = (const float*)d_in[7];
  const float* ad2 = (const float*)d_in[8];
  const float* b2  = (const float*)d_in[9];
  const float* W3  = (const float*)d_in[10];
  const float* b3  = (const float*)d_in[11];
  const float* W4  = (const float*)d_in[12];
  const float* b4  = (const float*)d_in[13];

  const int H1 = 4, C1 = 128, F1 = H1 * C1;   // layer-1: 4 heads x 128 = 512 (concat)
  const int C2 = 128;                          // layer-2: 1 head x 128
  const int F3 = 64, OD = 4;
  const int IN = in_sizes[2] / F1;             // 518
  const int N  = in_sizes[0] / IN;             // 50000
  const int E  = in_sizes[1] / 2;              // 800000
  const int ET = E + N;                        // + self loops
  const int* srcA = ei;
  const int* dstA = ei + E;

  // ---- workspace carve (256B aligned slabs) ----
  char* ws = (char*)d_ws;
  size_t off = 0;
  auto carve = [&](size_t bytes) -> void* {
    void* p = ws + off;
    off += (bytes + 255) & ~(size_t)255;
    return p;
  };
  float*    h1   = (float*)   carve((size_t)N * F1 * 4);  // GEMM1 out, later act1
  float*    agg1 = (float*)   carve((size_t)N * F1 * 4);
  float*    alsb1= (float*)   carve((size_t)N * H1 * 4);
  float*    aldb1= (float*)   carve((size_t)N * H1 * 4);
  unsigned* m1   = (unsigned*)carve((size_t)N * H1 * 4);
  float*    den1 = (float*)   carve((size_t)N * H1 * 4);
  float*    eb1  = (float*)   carve((size_t)ET * H1 * 4);
  float*    h2   = (float*)   carve((size_t)N * C2 * 4);  // GEMM2 out, later r2
  float*    agg2 = (float*)   carve((size_t)N * C2 * 4);
  float*    alsb2= (float*)   carve((size_t)N * 4);
  float*    aldb2= (float*)   carve((size_t)N * 4);
  unsigned* m2   = (unsigned*)carve((size_t)N * 4);
  float*    den2 = (float*)   carve((size_t)N * 4);
  float*    eb2  = (float*)   carve((size_t)ET * 4);
  float*    h3   = (float*)   carve((size_t)N * F3 * 4);
  (void)ws_size; (void)n_in; (void)out_size;

  auto zero = [&](void* p, size_t nwords) {
    int blocks = (int)((nwords + 255) / 256);
    if (blocks > 65535) blocks = 65535;
    fill_zero_kernel<<<dim3(blocks), dim3(256), 0, stream>>>((float*)p, nwords);
  };
  // zero for segment reductions (ordered-uint 0 == minimum, safe for max-init)
  zero(agg1, (size_t)N * F1);
  zero(m1,   (size_t)N * H1);
  zero(den1, (size_t)N * H1);
  zero(agg2, (size_t)N * C2);
  zero(m2,   (size_t)N);
  zero(den2, (size_t)N);

  const int TPB = 256, WPB = TPB / 32;

  // ===== Layer 1: GATConv(518 -> 4x128, concat) =====
  gemm_f32_wmma_kernel<<<dim3(N / 16, (F1 / 16 + WPB - 1) / WPB), dim3(TPB), 0, stream>>>(
      x, W1, h1, N, F1, IN);
  attn_scores_kernel<<<dim3((N * H1 + WPB - 1) / WPB), dim3(TPB), 0, stream>>>(
      h1, as1, ad1, alsb1, aldb1, N, H1, C1);
  edge_score_max_kernel<<<dim3((ET * H1 + TPB - 1) / TPB), dim3(TPB), 0, stream>>>(
      srcA, dstA, alsb1, aldb1, eb1, m1, E, ET, H1);
  edge_exp_sum_kernel<<<dim3((ET * H1 + TPB - 1) / TPB), dim3(TPB), 0, stream>>>(
      dstA, eb1, m1, den1, E, ET, H1);
  edge_aggregate_kernel<<<dim3((ET + WPB - 1) / WPB), dim3(TPB), 0, stream>>>(
      srcA, dstA, h1, eb1, den1, agg1, E, ET, H1, 7);
  bias_act_kernel<<<dim3(4096), dim3(TPB), 0, stream>>>(   // ELU(agg1 + b1) -> h1
      agg1, b1, h1, (size_t)N * F1, F1, 0);

  // ===== Layer 2: GATConv(512 -> 1x128, mean == identity for 1 head) =====
  gemm_f32_wmma_kernel<<<dim3(N / 16, (C2 / 16 + WPB - 1) / WPB), dim3(TPB), 0, stream>>>(
      h1, W2, h2, N, C2, F1);
  attn_scores_kernel<<<dim3((N + WPB - 1) / WPB), dim3(TPB), 0, stream>>>(
      h2, as2, ad2, alsb2, aldb2, N, 1, C2);
  edge_score_max_kernel<<<dim3((ET + TPB - 1) / TPB), dim3(TPB), 0, stream>>>(
      srcA, dstA, alsb2, aldb2, eb2, m2, E, ET, 1);
  edge_exp_sum_kernel<<<dim3((ET + TPB - 1) / TPB), dim3(TPB), 0, stream>>>(
      dstA, eb2, m2, den2, E, ET, 1);
  edge_aggregate_kernel<<<dim3((ET + WPB - 1) / WPB), dim3(TPB), 0, stream>>>(
      srcA, dstA, h2, eb2, den2, agg2, E, ET, 1, 7);
  bias_act_kernel<<<dim3(4096), dim3(TPB), 0, stream>>>(   // ReLU(agg2 + b2) -> h2
      agg2, b2, h2, (size_t)N * C2, C2, 1);

  // ===== MLP head =====
  gemm_f32_wmma_kernel<<<dim3(N / 16, (F3 / 16 + WPB - 1) / WPB), dim3(TPB), 0, stream>>>(
      h2, W3, h3, N, F3, C2);
  bias_act_kernel<<<dim3(4096), dim3(TPB), 0, stream>>>(   // ReLU(h3 + b3) in place
      h3, b3, h3, (size_t)N * F3, F3, 1);
  head_out_kernel<<<dim3((N * OD + TPB - 1) / TPB), dim3(TPB), 0, stream>>>(
      h3, W4, b4, (float*)d_out, N, F3, OD);
}